// GraphLearner_34608846471217
// MI455X (gfx1250) — compile-verified
//
#include <hip/hip_runtime.h>
#include <hip/hip_bf16.h>

// Problem constants (from reference): b=8, n=2048, f=256, heads=3, d=10
#define BB 8
#define NN 2048
#define FF 256
#define HH 3
#define DD 10
#define KP 12            // d padded to 12 so K splits into 3 chunks of 4 (zero pad)
#define NT (NN / 16)     // 128 tiles per dimension

typedef __attribute__((ext_vector_type(2))) float v2f;
typedef __attribute__((ext_vector_type(8))) float v8f;

// ---- Kernel 0: b_z passthrough copy (16 MB, float4) --------------------------
__global__ void k_copy_bz(const float4* __restrict__ src, float4* __restrict__ dst) {
    int i = blockIdx.x * blockDim.x + threadIdx.x;   // exactly n4 threads launched
    dst[i] = src[i];
}

// ---- Kernel 1: h[r, k] = leaky_relu( x[bn,:] . W[head,:,k] ), zero-padded ----
// r = (head*BB + b)*NN + n, k in [0, KP)
__global__ void k_project(const float* __restrict__ x, const float* __restrict__ W,
                          float* __restrict__ hbuf) {
    int idx = blockIdx.x * blockDim.x + threadIdx.x;   // r*KP + k, exact count
    int k = idx % KP;
    int r = idx / KP;
    int head = r / (BB * NN);
    int bn   = r % (BB * NN);
    float acc = 0.0f;
    if (k < DD) {
        const float* xr = x + (size_t)bn * FF;
        const float* wc = W + (size_t)head * FF * DD + k;   // stride DD over f
#pragma unroll 8
        for (int f = 0; f < FF; ++f) acc = fmaf(xr[f], wc[(size_t)f * DD], acc);
        acc = acc > 0.0f ? acc : 0.01f * acc;               // LeakyReLU(0.01)
    }
    hbuf[idx] = acc;                                        // k>=10 -> 0 pad
}

// ---- Kernel 2: reciprocal row norms -----------------------------------------
// rinv[r] = 1/||h[r,:]||.  1/max(nr*nc, eps) == min(rinv_r*rinv_c, 1/eps) for
// non-negative norms (0-norm -> inf -> clamps to 1/eps, matching the reference).
__global__ void k_norm(const float* __restrict__ hbuf, float* __restrict__ rnrm) {
    int r = blockIdx.x * blockDim.x + threadIdx.x;          // exact count
    const float* hr = hbuf + (size_t)r * KP;
    float s = 0.0f;
#pragma unroll
    for (int k = 0; k < KP; ++k) s = fmaf(hr[k], hr[k], s); // pads are zero
    rnrm[r] = 1.0f / sqrtf(s);
}

// ---- Kernel 3: WMMA tile kernel ---------------------------------------------
// One wave per 16x16 output tile of one batch; loops over 3 heads, writes
// b_adj_ per head and the mean b_adj. Grid sized exactly -> EXEC all ones.
__global__ void __launch_bounds__(256)
k_adj(const float* __restrict__ hbuf, const float* __restrict__ rnrm,
      float* __restrict__ adj_heads,   // [HH, BB, NN, NN]
      float* __restrict__ adj_mean) {  // [BB, NN, NN]
    const int lane = threadIdx.x & 31;
    const int tile = blockIdx.x * 8 + (threadIdx.x >> 5);   // 8 waves / block
    const int b  = tile / (NT * NT);
    const int t  = tile % (NT * NT);
    const int ti = t / NT;
    const int tj = t % NT;
    const int rowBase = ti * 16;
    const int colBase = tj * 16;

    const int lrow  = lane & 15;         // M (for A) / N (for B) per ISA layout
    const int khalf = (lane >> 4) * 2;   // K sub-offset per half-wave
    const int rhalf = (lane >> 4) * 8;   // C/D: M = v + 8*(lane/16)

    float mean_acc[8];
#pragma unroll
    for (int v = 0; v < 8; ++v) mean_acc[v] = 0.0f;

#pragma unroll
    for (int head = 0; head < HH; ++head) {
        const size_t rbase = (size_t)(head * BB + b) * NN * KP;
        const float* Arow = hbuf + rbase + (size_t)(rowBase + lrow) * KP + khalf;
        const float* Brow = hbuf + rbase + (size_t)(colBase + lrow) * KP + khalf;

        v8f c = {0.f, 0.f, 0.f, 0.f, 0.f, 0.f, 0.f, 0.f};
#pragma unroll
        for (int k0 = 0; k0 < KP; k0 += 4) {
            v2f a  = *(const v2f*)(Arow + k0);   // 8B-aligned: KP,k0,khalf all even
            v2f bm = *(const v2f*)(Brow + k0);
            // D = A(16x4,f32) * B(4x16,f32) + C  -> v_wmma_f32_16x16x4_f32
            c = __builtin_amdgcn_wmma_f32_16x16x4_f32(
                    false, a, false, bm, (short)0, c, false, false);
        }

        const size_t nbase = (size_t)(head * BB + b) * NN;
        const float  rcol  = rnrm[nbase + colBase + lrow];
        float* outh = adj_heads + ((size_t)(head * BB + b) * NN) * NN;

#pragma unroll
        for (int v = 0; v < 8; ++v) {
            const int row = rowBase + rhalf + v;
            const float rr    = rnrm[nbase + row];
            const float scale = fminf(rr * rcol, 1000.0f);  // == 1/max(nr*nc, 1e-3)
            float a = fabsf(c[v]) * scale;
            a = (a > 0.1f) ? a : 0.0f;                      // nn.Threshold(0.1, 0)
            mean_acc[v] += a;
            outh[(size_t)row * NN + colBase + lrow] = a;
        }
    }

    float* outm = adj_mean + (size_t)b * NN * NN;
#pragma unroll
    for (int v = 0; v < 8; ++v) {
        const int row = rowBase + rhalf + v;
        outm[(size_t)row * NN + colBase + lrow] = mean_acc[v] * (1.0f / 3.0f);
    }
}

// ---- Host launch ------------------------------------------------------------
extern "C" void kernel_launch(void* const* d_in, const int* in_sizes, int n_in,
                              void* d_out, int out_size, void* d_ws, size_t ws_size,
                              hipStream_t stream) {
    const float* b_z     = (const float*)d_in[0];   // [8,2048,256]
    const float* b_het_z = (const float*)d_in[1];   // [8,2048,256]
    const float* W       = (const float*)d_in[2];   // [3,256,10]

    float* out = (float*)d_out;
    float* out_bz      = out;                                   // 8*2048*256
    float* out_adj     = out + (size_t)BB * NN * FF;            // 8*2048*2048
    float* out_adj_hds = out_adj + (size_t)BB * NN * NN;        // 3*8*2048*2048

    // workspace: hbuf [H*B*N, KP] + reciprocal norms [H*B*N]  (~2.4 MB)
    float* hbuf = (float*)d_ws;
    float* rnrm = hbuf + (size_t)HH * BB * NN * KP;

    const int nrows = HH * BB * NN;                 // 49152

    // 0) b_z passthrough (float4): 1,048,576 vec4 elements
    k_copy_bz<<<(BB * NN * FF / 4) / 256, 256, 0, stream>>>(
        (const float4*)b_z, (float4*)out_bz);

    // 1) projection + LeakyReLU into padded h: nrows*KP = 589,824 threads
    k_project<<<(nrows * KP) / 256, 256, 0, stream>>>(b_het_z, W, hbuf);

    // 2) reciprocal row norms: 49,152 threads
    k_norm<<<nrows / 256, 256, 0, stream>>>(hbuf, rnrm);

    // 3) WMMA adjacency: B * NT * NT = 131,072 tiles, 8 waves/block
    k_adj<<<(BB * NT * NT) / 8, 256, 0, stream>>>(hbuf, rnrm, out_adj_hds, out_adj);
}